// EncodeLayer_56650618634690
// MI455X (gfx1250) — compile-verified
//
#include <hip/hip_runtime.h>
#include <cstdint>

// EncodeLayer for gfx1250 (MI455X).
// B=16 batches, N=512 labels, A=8732 anchors.
// labels: (B, N, 5) f32 [x1,y1,x2,y2,cls]; anchors: (A, 4) f32 [x1,y1,x2,y2]
// out: (B, A, 5) f32 [ox, oy, ow, oh, cls*pos]
//
// CDNA5-specific paths:
//  - global_load_async_to_lds_b128 (ASYNCcnt data mover) stages the 10 KB
//    label block into LDS once per workgroup
//  - s_wait_asynccnt 0 before the barrier
//  - inner loop reads each label as ONE ds_load_b128 broadcast (all 32 lanes
//    of a wave32 read the same address -> conflict-free broadcast)

#define NLAB 512
#define BLK  256

__device__ __forceinline__ void async_cp16(uint32_t lds_off, uint64_t gaddr) {
    asm volatile("global_load_async_to_lds_b128 %0, %1, off"
                 :: "v"(lds_off), "v"(gaddr)
                 : "memory");
}

__global__ __launch_bounds__(BLK) void encode_layer_kernel(
    const float* __restrict__ labels,
    const float* __restrict__ anchors,
    float* __restrict__ out,
    int B, int A)
{
    __shared__ float  slab[NLAB * 5];  // 10240 B: raw async-copied labels[b]
    __shared__ float4 sbox[NLAB];      //  8192 B: repacked corners (16B aligned)
    __shared__ float  scls[NLAB];      //  2048 B: class ids

    const int tid = threadIdx.x;
    const int b   = blockIdx.y;
    const int a   = blockIdx.x * BLK + tid;

    // ---- async DMA labels[b] (2560 dwords = 640 x b128) into LDS ----
    // 640 chunks over 256 threads: 2 full rounds + half round (tid < 128).
    {
        const uint64_t g0 = (uint64_t)(uintptr_t)(labels + (size_t)b * NLAB * 5);
        const uint32_t l0 = (uint32_t)(uintptr_t)(&slab[0]);  // low 32b of generic = LDS offset
        async_cp16(l0 + (uint32_t)(tid * 16u),          g0 + (uint64_t)(tid * 16u));
        async_cp16(l0 + (uint32_t)((tid + 256) * 16u),  g0 + (uint64_t)((tid + 256) * 16u));
        if (tid < 128)
            async_cp16(l0 + (uint32_t)((tid + 512) * 16u), g0 + (uint64_t)((tid + 512) * 16u));
    }
    asm volatile("s_wait_asynccnt 0x0" ::: "memory");
    __syncthreads();

    // ---- repack AoS(stride 5) -> float4 corners + cls for b128 broadcast ----
    for (int j = tid; j < NLAB; j += BLK) {
        sbox[j] = make_float4(slab[j * 5 + 0], slab[j * 5 + 1],
                              slab[j * 5 + 2], slab[j * 5 + 3]);
        scls[j] = slab[j * 5 + 4];
    }
    __syncthreads();

    if (a >= A) return;

    // anchor corners (coalesced 16B load)
    const float4 an    = ((const float4*)anchors)[a];   // x1 y1 x2 y2
    const float  aarea = an.z * an.w;                   // reference quirk: area = x2*y2

    // division-free running argmax of IoU over 512 labels
    float best_inter = -1.0f;
    float best_denom =  1.0f;
    int   best       =  0;

#pragma unroll 4
    for (int j = 0; j < NLAB; ++j) {
        const float4 L = sbox[j];                       // one ds_load_b128 broadcast

        float iw = fminf(L.z, an.z) - fmaxf(L.x, an.x);
        float ih = fminf(L.w, an.w) - fmaxf(L.y, an.y);
        iw = fmaxf(iw, 0.0f);
        ih = fmaxf(ih, 0.0f);
        const float inter = iw * ih;
        const float denom = fmaf(L.z, L.w, aarea - inter);  // area1 + area2 - inter, > 0

        // iou_j > iou_best  <=>  inter*best_denom > best_inter*denom (denoms > 0)
        // strict '>' keeps the first (lowest) index on ties, matching jnp.argmax
        if (inter * best_denom > best_inter * denom) {
            best_inter = inter;
            best_denom = denom;
            best       = j;
        }
    }

    // gather matched label (divergent LDS read, once per thread)
    const float4 m   = sbox[best];
    const float  mcl = scls[best];

    // corner -> cxcywh for both boxes
    const float tcx = 0.5f * (m.x + m.z), tcy = 0.5f * (m.y + m.w);
    const float tw  = m.z - m.x,          th  = m.w - m.y;
    const float acx = 0.5f * (an.x + an.z), acy = 0.5f * (an.y + an.w);
    const float aw  = an.z - an.x,          ah  = an.w - an.y;

    const float inv_aw = 1.0f / aw;
    const float inv_ah = 1.0f / ah;
    const float ox = (tcx - acx) * inv_aw;
    const float oy = (tcy - acy) * inv_ah;
    const float ow = __logf(tw * inv_aw);
    const float oh = __logf(th * inv_ah);

    // pos = (iou_max >= 0.5) without dividing
    const float pos = (best_inter >= 0.5f * best_denom) ? 1.0f : 0.0f;
    const float cls = mcl * pos;

    float* o = out + ((size_t)b * A + a) * 5;
    o[0] = ox;
    o[1] = oy;
    o[2] = ow;
    o[3] = oh;
    o[4] = cls;
}

extern "C" void kernel_launch(void* const* d_in, const int* in_sizes, int n_in,
                              void* d_out, int out_size, void* d_ws, size_t ws_size,
                              hipStream_t stream) {
    const float* labels  = (const float*)d_in[0];  // (B, N, 5) f32
    const float* anchors = (const float*)d_in[1];  // (A, 4)    f32
    float*       out     = (float*)d_out;          // (B, A, 5) f32

    const int A = in_sizes[1] / 4;          // 8732
    const int B = in_sizes[0] / (NLAB * 5); // 16

    dim3 grid((A + BLK - 1) / BLK, B);
    encode_layer_kernel<<<grid, BLK, 0, stream>>>(labels, anchors, out, B, A);
}